// L2_20701742367347
// MI455X (gfx1250) — compile-verified
//
#include <hip/hip_runtime.h>

// CDNA5 / gfx1250: wave32, WMMA f32 16x16x4 used as a row-reduction engine.
typedef float v2f __attribute__((ext_vector_type(2)));
typedef float v8f __attribute__((ext_vector_type(8)));

#define DIM      256
#define L1_DIM   128
#define WAVES_PB 8   // 256 threads / 32

// One wave32 processes a tile of 16 edges.
// A (16x4 f32): lane l & l+16 hold row M=l; VGPR0/1 = K0,K1 (lanes 0-15) or K2,K3 (lanes 16-31).
// B = all-ones 4x16 -> every column of the 16x16 f32 accumulator D is the row sum.
__global__ __launch_bounds__(256) void edge_mu_var_wmma(
    const float* __restrict__ lig,
    const float* __restrict__ prot,
    const int*   __restrict__ eidx,   // [2, E] int32 (JAX x64-off)
    float*       __restrict__ out,    // [E, 2] = (mu, var)
    int E)
{
    const int lane = threadIdx.x & 31;
    const int wave = threadIdx.x >> 5;
    const long base = ((long)blockIdx.x * WAVES_PB + wave) * 16;
    if (base >= E) return;                   // whole-wave uniform exit only

    // Edge id per lane (lanes l and l+16 map to the same matrix row M = l&15).
    const int m = lane & 15;
    long e = base + m;
    if (e >= E) e = E - 1;                   // per-lane clamp, EXEC stays all-ones
    const int s = eidx[e];                   // ligand row
    const int d = eidx[(long)E + e];         // protein row

    // Lane-half K offset inside each 4-wide chunk: lanes 16-31 take K=2,3 (+8 bytes).
    const int halfOfs = (lane >> 4) << 1;    // 0 or 2 floats
    const v2f* __restrict__ lrow = (const v2f*)(lig  + (long)s * DIM + halfOfs);
    const v2f* __restrict__ prow = (const v2f*)(prot + (long)d * DIM + halfOfs);

    __builtin_prefetch(lrow, 0, 3);          // global_prefetch_b8 — warm gathered rows
    __builtin_prefetch(prow, 0, 3);

    v8f cv = {0.f,0.f,0.f,0.f,0.f,0.f,0.f,0.f};   // L1 (abs) accumulator
    v8f cm = {0.f,0.f,0.f,0.f,0.f,0.f,0.f,0.f};   // L2 (square) accumulator
    const v2f ones = {1.0f, 1.0f};                // B matrix: all ones (layout-independent)

    // Channels 0..127: var += |lig - prot|, reduced across K by the matrix pipe.
#pragma unroll
    for (int k = 0; k < L1_DIM / 4; ++k) {
        v2f a = lrow[2 * k];                 // global_load_b64, static offset k*16
        v2f b = prow[2 * k];
        v2f A;
        A.x = __builtin_fabsf(a.x - b.x);
        A.y = __builtin_fabsf(a.y - b.y);
        cv = __builtin_amdgcn_wmma_f32_16x16x4_f32(
                 false, A, false, ones, (short)0, cv, false, false);
    }
    // Channels 128..255: mu2 += (lig - prot)^2.
#pragma unroll
    for (int k = L1_DIM / 4; k < DIM / 4; ++k) {
        v2f a = lrow[2 * k];
        v2f b = prow[2 * k];
        const float d0 = a.x - b.x;
        const float d1 = a.y - b.y;
        v2f A = { d0 * d0, d1 * d1 };
        cm = __builtin_amdgcn_wmma_f32_16x16x4_f32(
                 false, A, false, ones, (short)0, cm, false, false);
    }

    // D layout: VGPR r -> M=r (lanes 0-15) / M=8+r (lanes 16-31); all N columns equal.
    // Bounce the 16 row-sums through per-wave LDS, then 16 lanes store (mu, var) pairs.
    __shared__ float sv[WAVES_PB][16];
    __shared__ float sm[WAVES_PB][16];
    if (lane == 0) {
#pragma unroll
        for (int r = 0; r < 8; ++r) { sv[wave][r] = cv[r]; sm[wave][r] = cm[r]; }
    } else if (lane == 16) {
#pragma unroll
        for (int r = 0; r < 8; ++r) { sv[wave][8 + r] = cv[r]; sm[wave][8 + r] = cm[r]; }
    }
    __syncthreads();                         // orders ds_store -> ds_load (s_wait_dscnt)

    if (lane < 16) {
        const long eo = base + lane;
        if (eo < E) {
            v2f o;
            o.x = sqrtf(sm[wave][lane]);     // mu
            o.y = sv[wave][lane];            // var
            *(v2f*)(out + 2 * eo) = o;       // global_store_b64, contiguous per tile
        }
    }
}

extern "C" void kernel_launch(void* const* d_in, const int* in_sizes, int n_in,
                              void* d_out, int out_size, void* d_ws, size_t ws_size,
                              hipStream_t stream) {
    const float* lig  = (const float*)d_in[0];
    const float* prot = (const float*)d_in[1];
    const int*   eidx = (const int*)d_in[2];
    const int E = in_sizes[2] / 2;           // edge_index is [2, E]
    float* out = (float*)d_out;

    const int tiles  = (E + 15) / 16;
    const int blocks = (tiles + WAVES_PB - 1) / WAVES_PB;
    edge_mu_var_wmma<<<blocks, 32 * WAVES_PB, 0, stream>>>(lig, prot, eidx, out, E);
}